// PolynomialBasis_30734785970297
// MI455X (gfx1250) — compile-verified
//
#include <hip/hip_runtime.h>

// Shifted-Legendre orthonormal basis on [0,1], degree 0..15.
// Strategy: per-lane Chebyshev recurrence (stable, |T_k|<=1) then a constant
// 16x16 Chebyshev->scaled-Legendre basis change done with chained
// V_WMMA_F32_16X16X4_F32 ops (exact-f32 WMMA path on gfx1250).
// The kernel is store-bandwidth bound (1 GiB out @ 23.3 TB/s ~= 47 us floor);
// WMMA rides along for free on otherwise-idle matrix pipes.

#define NB 16  // DEGREE+1

typedef float v2f __attribute__((ext_vector_type(2)));
typedef float v8f __attribute__((ext_vector_type(8)));

struct Mat { float v[NB][NB]; };

// constexpr sqrt (Newton), since std::sqrt isn't constexpr here
constexpr double csqrt_(double x) {
    double r = x < 1.0 ? 1.0 : x;
    for (int i = 0; i < 64; ++i) r = 0.5 * (r + x / r);
    return r;
}

// Compile-time: coefficients of sqrt(2i+1)*P_i expressed in Chebyshev T_k,
// via Bonnet recurrence on Chebyshev-coefficient vectors:
//   x*T_0 = T_1 ;  x*T_k = (T_{k+1} + T_{k-1})/2
constexpr Mat make_coef() {
    double P[NB][NB + 2] = {};   // P[i][k] = coeff of T_k in Legendre P_i
    P[0][0] = 1.0;
    P[1][1] = 1.0;
    for (int i = 1; i < NB - 1; ++i) {
        double xP[NB + 2] = {};
        for (int k = 0; k <= i; ++k) {
            double c = P[i][k];
            if (k == 0) { xP[1] += c; }
            else        { xP[k - 1] += 0.5 * c; xP[k + 1] += 0.5 * c; }
        }
        for (int k = 0; k <= i + 1; ++k)
            P[i + 1][k] = ((2.0 * i + 1.0) * xP[k] - (double)i * P[i - 1][k]) / (double)(i + 1);
    }
    Mat m = {};
    for (int i = 0; i < NB; ++i) {
        double s = csqrt_(2.0 * i + 1.0);
        for (int k = 0; k < NB; ++k) m.v[i][k] = (float)(s * P[i][k]);
    }
    return m;
}

__device__ __constant__ Mat g_coef = make_coef();

__global__ __launch_bounds__(256) void PolynomialBasis_30734785970297_kernel(
    const float* __restrict__ uin, float* __restrict__ out, long long ntiles)
{
    const int lane  = threadIdx.x & 31;
    const int wave  = threadIdx.x >> 5;
    const long long gwave  = (long long)blockIdx.x * (blockDim.x >> 5) + wave;
    const long long nwaves = (long long)gridDim.x * (blockDim.x >> 5);

    const int  row = lane & 15;      // point index within tile / A-matrix row
    const bool kh  = lane >= 16;     // which half of the K-range this lane holds

    // Loop-invariant A-matrix in WMMA 16x4 f32 layout:
    // lanes 0-15: K = {4j, 4j+1}; lanes 16-31: K = {4j+2, 4j+3}
    const float* cr = &g_coef.v[row][0];
    const int ko = kh ? 2 : 0;
    const v2f a0 = { cr[0 + ko],  cr[1 + ko]  };
    const v2f a1 = { cr[4 + ko],  cr[5 + ko]  };
    const v2f a2 = { cr[8 + ko],  cr[9 + ko]  };
    const v2f a3 = { cr[12 + ko], cr[13 + ko] };

    for (long long t = gwave; t < ntiles; t += nwaves) {   // wave-uniform loop
        const long long p = (t << 4) + row;                // this lane's point
        const float x  = fmaf(2.0f, uin[p], -1.0f);        // [0,1] -> [-1,1]
        const float x2 = x + x;

        // Chebyshev recurrence: stable, fully in registers
        float T[16];
        T[0] = 1.0f; T[1] = x;
#pragma unroll
        for (int k = 2; k < 16; ++k) T[k] = fmaf(x2, T[k - 1], -T[k - 2]);

        // B-matrix (4x16 f32, 2 VGPRs) per K-chunk: same half-split as A
        const v2f b0 = { kh ? T[2]  : T[0],  kh ? T[3]  : T[1]  };
        const v2f b1 = { kh ? T[6]  : T[4],  kh ? T[7]  : T[5]  };
        const v2f b2 = { kh ? T[10] : T[8],  kh ? T[11] : T[9]  };
        const v2f b3 = { kh ? T[14] : T[12], kh ? T[15] : T[13] };

        // D(16x16) = Coef(16x16) x ChebVals(16x16), chained K=4 accumulation
        v8f c = {};
        c = __builtin_amdgcn_wmma_f32_16x16x4_f32(false, a0, false, b0, (short)0, c, false, false);
        c = __builtin_amdgcn_wmma_f32_16x16x4_f32(false, a1, false, b1, (short)0, c, false, false);
        c = __builtin_amdgcn_wmma_f32_16x16x4_f32(false, a2, false, b2, (short)0, c, false, false);
        c = __builtin_amdgcn_wmma_f32_16x16x4_f32(false, a3, false, b3, (short)0, c, false, false);

        // D layout: vgpr r = basis (r + 8*kh) of point p -> two b128 stores/lane,
        // wave covers a contiguous, fully-written 1 KiB span per tile.
        float* op = out + (p << 4) + (kh ? 8 : 0);
        *(float4*)(op)     = make_float4(c[0], c[1], c[2], c[3]);
        *(float4*)(op + 4) = make_float4(c[4], c[5], c[6], c[7]);
    }
}

extern "C" void kernel_launch(void* const* d_in, const int* in_sizes, int n_in,
                              void* d_out, int out_size, void* d_ws, size_t ws_size,
                              hipStream_t stream) {
    const float* u = (const float*)d_in[0];
    float* out = (float*)d_out;
    const long long n = in_sizes[0];
    const long long ntiles = n >> 4;          // 16 points per WMMA tile

    const int threads = 256;                  // 8 wave32 per block
    long long maxBlocks = (ntiles + 7) / 8;   // at least one tile per wave
    int blocks = 4096;
    if ((long long)blocks > maxBlocks) blocks = (int)maxBlocks;
    if (blocks < 1) blocks = 1;

    PolynomialBasis_30734785970297_kernel<<<blocks, threads, 0, stream>>>(u, out, ntiles);
}